// MMD_loss_69148973465980
// MI455X (gfx1250) — compile-verified
//
#include <hip/hip_runtime.h>
#include <hip/hip_bf16.h>
#include <math.h>

typedef __attribute__((ext_vector_type(2))) float v2f;
typedef __attribute__((ext_vector_type(8))) float v8f;

#define NHALF 2048
#define NTOT  4096
#define DDIM  1024
#define NELEM (16777216ull)          // 4096*4096
#define KRANK 8388607u               // (16777216-1)/2

struct SelState {
    unsigned prefix;
    unsigned krem;
    float    scale;   // 1/(2*bw^2)
    float    bw;
};

__device__ __forceinline__ const float* row_ptr(const float* src, const float* tgt, int r) {
    return (r < NHALF) ? (src + (size_t)r * DDIM) : (tgt + (size_t)(r - NHALF) * DDIM);
}

// ---------------- init: zero hist / state / accumulator ----------------
__global__ void init_kernel(unsigned* hist, SelState* st, double* acc) {
    for (int i = threadIdx.x; i < 2048; i += blockDim.x) hist[i] = 0u;
    if (threadIdx.x == 0) {
        st->prefix = 0u;
        st->krem   = KRANK;
        st->scale  = 0.f;
        st->bw     = 0.f;
        *acc       = 0.0;
    }
}

// ---------------- per-row squared norms ----------------
__global__ __launch_bounds__(256) void sq_kernel(const float* __restrict__ src,
                                                 const float* __restrict__ tgt,
                                                 float* __restrict__ sq) {
    int row = blockIdx.x;
    const float* p = row_ptr(src, tgt, row);
    float s = 0.f;
    for (int c = threadIdx.x; c < DDIM; c += 256) {
        float v = p[c];
        s += v * v;
    }
    __shared__ float red[256];
    red[threadIdx.x] = s;
    __syncthreads();
    for (int off = 128; off > 0; off >>= 1) {
        if (threadIdx.x < off) red[threadIdx.x] += red[threadIdx.x + off];
        __syncthreads();
    }
    if (threadIdx.x == 0) sq[row] = red[0];
}

// ---------------- WMMA f32 GEMM -> L2 matrix ----------------
// Wave computes a 32x32 output tile via 2x2 fragments of V_WMMA_F32_16X16X4_F32.
// Block = 8 waves (4x2) -> 128x64 block tile. Grid = (4096/64, 4096/128).
__global__ __launch_bounds__(256) void gemm_l2_kernel(const float* __restrict__ src,
                                                      const float* __restrict__ tgt,
                                                      const float* __restrict__ sq,
                                                      float* __restrict__ L2m) {
    const int lane  = threadIdx.x & 31;
    const int w     = threadIdx.x >> 5;       // 0..7
    const int wi    = w >> 1;                 // 0..3
    const int wj    = w & 1;                  // 0..1
    const int i0    = blockIdx.y * 128 + wi * 32;
    const int j0    = blockIdx.x * 64  + wj * 32;
    const int laneM = lane & 15;
    const int koff  = (lane >> 4) * 2;        // K sub-offset per lane half

    // A fragment rows (M dim), B fragment rows (N dim of output, rows of total)
    const float* pA0 = row_ptr(src, tgt, i0 + laneM)      + koff;
    const float* pA1 = row_ptr(src, tgt, i0 + 16 + laneM) + koff;
    const float* pB0 = row_ptr(src, tgt, j0 + laneM)      + koff;
    const float* pB1 = row_ptr(src, tgt, j0 + 16 + laneM) + koff;

    v8f c00 = {}, c01 = {}, c10 = {}, c11 = {};

    for (int kk = 0; kk < DDIM; kk += 4) {
        v2f a0 = *(const v2f*)(pA0 + kk);
        v2f a1 = *(const v2f*)(pA1 + kk);
        v2f b0 = *(const v2f*)(pB0 + kk);
        v2f b1 = *(const v2f*)(pB1 + kk);
        c00 = __builtin_amdgcn_wmma_f32_16x16x4_f32(false, a0, false, b0, (short)0, c00, false, false);
        c01 = __builtin_amdgcn_wmma_f32_16x16x4_f32(false, a0, false, b1, (short)0, c01, false, false);
        c10 = __builtin_amdgcn_wmma_f32_16x16x4_f32(false, a1, false, b0, (short)0, c10, false, false);
        c11 = __builtin_amdgcn_wmma_f32_16x16x4_f32(false, a1, false, b1, (short)0, c11, false, false);
    }

    // Epilogue: L2 = max(sq_i + sq_j - 2*gram, 0)
    const int mrow = (lane >> 4) * 8;   // C layout: lanes 16-31 hold M+8
    const int ncol = lane & 15;
#pragma unroll
    for (int mi = 0; mi < 2; ++mi) {
#pragma unroll
        for (int mj = 0; mj < 2; ++mj) {
            v8f cc = (mi == 0) ? ((mj == 0) ? c00 : c01) : ((mj == 0) ? c10 : c11);
#pragma unroll
            for (int r = 0; r < 8; ++r) {
                int row = i0 + mi * 16 + mrow + r;
                int col = j0 + mj * 16 + ncol;
                float l2 = sq[row] + sq[col] - 2.0f * cc[r];
                L2m[(size_t)row * NTOT + col] = fmaxf(l2, 0.0f);
            }
        }
    }
}

// ---------------- radix-select histogram pass ----------------
// pass 0: bin = bits >> 21 (top 11)
// pass 1: among bits whose top 11 == prefix: bin = (bits >> 10) & 0x7FF
// pass 2: among bits whose top 22 == prefix: bin = bits & 0x3FF
__global__ __launch_bounds__(256) void hist_kernel(const float* __restrict__ L2m,
                                                   const SelState* __restrict__ st,
                                                   unsigned* __restrict__ hist,
                                                   int pass) {
    __shared__ unsigned lh[2048];
    for (int i = threadIdx.x; i < 2048; i += 256) lh[i] = 0u;
    __syncthreads();
    const unsigned prefix = st->prefix;
    size_t base = (size_t)blockIdx.x * 4096 + threadIdx.x;
    for (int t = 0; t < 16; ++t) {
        size_t idx = base + (size_t)t * 256;
        unsigned v = __float_as_uint(L2m[idx]);
        bool match;
        unsigned bin;
        if (pass == 0)      { match = true;                      bin = v >> 21; }
        else if (pass == 1) { match = ((v >> 21) == prefix);     bin = (v >> 10) & 2047u; }
        else                { match = ((v >> 10) == prefix);     bin = v & 1023u; }
        if (match) atomicAdd(&lh[bin], 1u);
    }
    __syncthreads();
    for (int i = threadIdx.x; i < 2048; i += 256) {
        unsigned c = lh[i];
        if (c) atomicAdd(&hist[i], c);
    }
}

// ---------------- scan histogram, update selection, zero hist ----------------
__global__ void scan_kernel(unsigned* __restrict__ hist,
                            SelState* __restrict__ st,
                            float* __restrict__ out,
                            int pass) {
    if (threadIdx.x == 0) {
        unsigned k = st->krem;
        unsigned cum = 0u, sel = 0u;
        for (unsigned b = 0; b < 2048u; ++b) {
            unsigned c = hist[b];
            if (cum + c > k) { sel = b; break; }
            cum += c;
        }
        st->krem = k - cum;
        if (pass == 0) {
            st->prefix = sel;
        } else if (pass == 1) {
            st->prefix = (st->prefix << 11) | sel;
        } else {
            unsigned bits = (st->prefix << 10) | sel;
            float median = __uint_as_float(bits);
            float bw = sqrtf(0.5f * median / logf(2048.0f + 1e-8f));
            st->bw = bw;
            st->scale = 1.0f / (2.0f * bw * bw);
            out[1] = bw;   // bandwidth output
        }
    }
    __syncthreads();
    for (int i = threadIdx.x; i < 2048; i += blockDim.x) hist[i] = 0u;
}

// ---------------- signed exp-kernel sum ----------------
__global__ __launch_bounds__(256) void loss_kernel(const float* __restrict__ L2m,
                                                   const SelState* __restrict__ st,
                                                   double* __restrict__ acc) {
    const float scale = st->scale;
    size_t base = (size_t)blockIdx.x * 4096 + threadIdx.x;
    float s = 0.f;
    for (int t = 0; t < 16; ++t) {
        size_t idx = base + (size_t)t * 256;
        unsigned row = (unsigned)(idx >> 12);
        unsigned col = (unsigned)(idx & 4095u);
        float v = expf(-L2m[idx] * scale);
        bool same = (row < NHALF) == (col < NHALF);
        s += same ? v : -v;
    }
    __shared__ float red[256];
    red[threadIdx.x] = s;
    __syncthreads();
    for (int off = 128; off > 0; off >>= 1) {
        if (threadIdx.x < off) red[threadIdx.x] += red[threadIdx.x + off];
        __syncthreads();
    }
    if (threadIdx.x == 0) atomicAdd(acc, (double)red[0]);
}

__global__ void finalize_kernel(const double* __restrict__ acc, float* __restrict__ out) {
    if (threadIdx.x == 0 && blockIdx.x == 0) {
        out[0] = (float)(*acc / 4194304.0);   // / N^2 (N=2048)
    }
}

extern "C" void kernel_launch(void* const* d_in, const int* in_sizes, int n_in,
                              void* d_out, int out_size, void* d_ws, size_t ws_size,
                              hipStream_t stream) {
    const float* src = (const float*)d_in[0];
    const float* tgt = (const float*)d_in[1];
    float* out = (float*)d_out;

    // workspace layout
    float*    L2m  = (float*)d_ws;                       // 4096*4096 f32 = 64 MiB
    float*    sq   = L2m + NELEM;                        // 4096 f32
    unsigned* hist = (unsigned*)(sq + NTOT);             // 2048 u32
    SelState* st   = (SelState*)(hist + 2048);           // 16 B
    double*   acc  = (double*)((char*)st + sizeof(SelState)); // 8 B, 8-aligned

    init_kernel<<<1, 256, 0, stream>>>(hist, st, acc);
    sq_kernel<<<NTOT, 256, 0, stream>>>(src, tgt, sq);
    gemm_l2_kernel<<<dim3(NTOT / 64, NTOT / 128), 256, 0, stream>>>(src, tgt, sq, L2m);

    for (int pass = 0; pass < 3; ++pass) {
        hist_kernel<<<4096, 256, 0, stream>>>(L2m, st, hist, pass);
        scan_kernel<<<1, 256, 0, stream>>>(hist, st, out, pass);
    }

    loss_kernel<<<4096, 256, 0, stream>>>(L2m, st, acc);
    finalize_kernel<<<1, 64, 0, stream>>>(acc, out);
}